// CSPFM_24713241821852
// MI455X (gfx1250) — compile-verified
//
#include <hip/hip_runtime.h>

typedef float v2f __attribute__((ext_vector_type(2)));
typedef float v4f __attribute__((ext_vector_type(4)));
typedef float v8f __attribute__((ext_vector_type(8)));

#define C_DIM 512
#define HW    4096   // 64*64
#define BATCH 32
#define NEG_BIG (-3.402823466e38f)

// ---------------------------------------------------------------------------
// Kernel 1: per-(b,c) global max + mean over H*W (4096 elems). One block per
// (b,c), 256 threads, each streaming 4 nontemporal float4 loads.
// ---------------------------------------------------------------------------
__global__ __launch_bounds__(256) void cspfm_pool_kernel(
    const float* __restrict__ x, float* __restrict__ d_out, float* __restrict__ e_out) {
  const int bc  = blockIdx.x;
  const int tid = threadIdx.x;
  const v4f* xv = reinterpret_cast<const v4f*>(x + (size_t)bc * HW);

  float vmax = NEG_BIG;
  float vsum = 0.0f;
#pragma unroll
  for (int k = 0; k < 4; ++k) {
    v4f v = __builtin_nontemporal_load(&xv[tid + k * 256]);
    vmax = fmaxf(vmax, fmaxf(fmaxf(v.x, v.y), fmaxf(v.z, v.w)));
    vsum += (v.x + v.y) + (v.z + v.w);
  }
  // wave32 reduction
#pragma unroll
  for (int m = 16; m >= 1; m >>= 1) {
    vmax = fmaxf(vmax, __shfl_xor(vmax, m, 32));
    vsum += __shfl_xor(vsum, m, 32);
  }
  __shared__ float smax[8];
  __shared__ float ssum[8];
  const int wave = tid >> 5;
  const int lane = tid & 31;
  if (lane == 0) { smax[wave] = vmax; ssum[wave] = vsum; }
  __syncthreads();
  if (tid == 0) {
    float m = smax[0], s = ssum[0];
#pragma unroll
    for (int i = 1; i < 8; ++i) { m = fmaxf(m, smax[i]); s += ssum[i]; }
    d_out[bc] = m;
    e_out[bc] = s * (1.0f / (float)HW);
  }
}

// ---------------------------------------------------------------------------
// Kernel 2: per-batch softmax(d⊗d + e⊗e) row-softmax + weighted column sums.
// One block (512 threads = 16 waves) per batch. WMMA f32 16x16x4 computes
// each 16x16 tile of the rank-2 Gram matrix (K slots: d, e, 0, 0).
// 3-pass streaming softmax per 16-row tile: rowmax -> rowsum -> accumulate
// f_max[j] += (d[i]/rowsum[i])*exp(.), f_mean[j] += (e[i]/rowsum[i])*exp(.).
// C/D layout: VGPR r holds row M=r (lanes 0-15) and M=r+8 (lanes 16-31).
// jt loops are kept rolled (unroll(disable)) so only ONE 8-VGPR accumulator
// tile is live at a time -> no scratch spills.
// ---------------------------------------------------------------------------
__global__ __launch_bounds__(512) void cspfm_attn_kernel(
    const float* __restrict__ d_g, const float* __restrict__ e_g,
    const float* __restrict__ alpha, const float* __restrict__ beta,
    float* __restrict__ f_g) {
  const int b    = blockIdx.x;
  const int tid  = threadIdx.x;
  const int lane = tid & 31;
  const int wave = tid >> 5;
  const bool hi  = (lane >= 16);
  const int l16  = lane & 15;

  __shared__ float ds[C_DIM];
  __shared__ float es[C_DIM];
  __shared__ float fmax_s[C_DIM];
  __shared__ float fmean_s[C_DIM];

  ds[tid]      = d_g[b * C_DIM + tid];
  es[tid]      = e_g[b * C_DIM + tid];
  fmax_s[tid]  = 0.0f;
  fmean_s[tid] = 0.0f;
  __syncthreads();

  // each wave owns i-tiles {wave, wave+16}  (32 tiles of 16 rows)
#pragma clang loop unroll(disable)
  for (int it = wave; it < 32; it += 16) {
    const int i0 = it * 16;

    // A (16x4, f32): lanes 0-15 carry K=0 (d) / K=1 (e); lanes 16-31 (K=2,3)=0
    const float dA = ds[i0 + l16];   // always in range; load unconditionally
    const float eA = es[i0 + l16];
    v2f A;
    A.x = hi ? 0.0f : dA;
    A.y = hi ? 0.0f : eA;

    // ---- pass 1: row maxima ----
    float rowmax[8];
#pragma unroll
    for (int r = 0; r < 8; ++r) rowmax[r] = NEG_BIG;
#pragma clang loop unroll(disable)
    for (int jt = 0; jt < 32; ++jt) {
      const float dB = ds[jt * 16 + l16];
      const float eB = es[jt * 16 + l16];
      v2f Bv;
      Bv.x = hi ? 0.0f : dB;
      Bv.y = hi ? 0.0f : eB;
      v8f acc = {};
      acc = __builtin_amdgcn_wmma_f32_16x16x4_f32(false, A, false, Bv,
                                                  (short)0, acc, false, false);
#pragma unroll
      for (int r = 0; r < 8; ++r) rowmax[r] = fmaxf(rowmax[r], acc[r]);
    }
#pragma unroll
    for (int r = 0; r < 8; ++r) {
#pragma unroll
      for (int m = 1; m < 16; m <<= 1)
        rowmax[r] = fmaxf(rowmax[r], __shfl_xor(rowmax[r], m, 32));
    }

    // ---- pass 2: row sums of exp ----
    float rowsum[8];
#pragma unroll
    for (int r = 0; r < 8; ++r) rowsum[r] = 0.0f;
#pragma clang loop unroll(disable)
    for (int jt = 0; jt < 32; ++jt) {
      const float dB = ds[jt * 16 + l16];
      const float eB = es[jt * 16 + l16];
      v2f Bv;
      Bv.x = hi ? 0.0f : dB;
      Bv.y = hi ? 0.0f : eB;
      v8f acc = {};
      acc = __builtin_amdgcn_wmma_f32_16x16x4_f32(false, A, false, Bv,
                                                  (short)0, acc, false, false);
#pragma unroll
      for (int r = 0; r < 8; ++r) rowsum[r] += __expf(acc[r] - rowmax[r]);
    }
#pragma unroll
    for (int r = 0; r < 8; ++r) {
#pragma unroll
      for (int m = 1; m < 16; m <<= 1)
        rowsum[r] += __shfl_xor(rowsum[r], m, 32);
    }

    // per-row weights d[i]/rowsum, e[i]/rowsum (lane-half selects row r / r+8)
    float wd[8], we[8];
#pragma unroll
    for (int r = 0; r < 8; ++r) {
      const int row   = i0 + r + (hi ? 8 : 0);
      const float inv = 1.0f / rowsum[r];
      wd[r] = ds[row] * inv;
      we[r] = es[row] * inv;
    }

    // ---- pass 3: weighted column accumulation ----
#pragma clang loop unroll(disable)
    for (int jt = 0; jt < 32; ++jt) {
      const float dB = ds[jt * 16 + l16];
      const float eB = es[jt * 16 + l16];
      v2f Bv;
      Bv.x = hi ? 0.0f : dB;
      Bv.y = hi ? 0.0f : eB;
      v8f acc = {};
      acc = __builtin_amdgcn_wmma_f32_16x16x4_f32(false, A, false, Bv,
                                                  (short)0, acc, false, false);
      float pmax = 0.0f, pmean = 0.0f;
#pragma unroll
      for (int r = 0; r < 8; ++r) {
        const float t = __expf(acc[r] - rowmax[r]);
        pmax  += wd[r] * t;
        pmean += we[r] * t;
      }
      // fold rows 8-15 half onto rows 0-7 half (same column)
      pmax  += __shfl_xor(pmax, 16, 32);
      pmean += __shfl_xor(pmean, 16, 32);
      if (!hi) {
        atomicAdd(&fmax_s[jt * 16 + l16], pmax);    // ds_add_f32
        atomicAdd(&fmean_s[jt * 16 + l16], pmean);  // ds_add_f32
      }
    }
  }
  __syncthreads();

  f_g[b * C_DIM + tid] = alpha[0] * fmax_s[tid] + beta[0] * fmean_s[tid];
}

// ---------------------------------------------------------------------------
// Kernel 3: broadcast f[b,c] over H*W with nontemporal float4 stores.
// ---------------------------------------------------------------------------
__global__ __launch_bounds__(256) void cspfm_bcast_kernel(
    const float* __restrict__ f, float* __restrict__ out) {
  const int bc  = blockIdx.x;
  const int tid = threadIdx.x;
  const float val = f[bc];
  v4f v = {val, val, val, val};
  v4f* o = reinterpret_cast<v4f*>(out + (size_t)bc * HW);
#pragma unroll
  for (int k = 0; k < 4; ++k)
    __builtin_nontemporal_store(v, &o[tid + k * 256]);
}

// ---------------------------------------------------------------------------
extern "C" void kernel_launch(void* const* d_in, const int* in_sizes, int n_in,
                              void* d_out, int out_size, void* d_ws, size_t ws_size,
                              hipStream_t stream) {
  const float* x     = (const float*)d_in[0];
  const float* alpha = (const float*)d_in[1];
  const float* beta  = (const float*)d_in[2];
  float* out = (float*)d_out;

  float* ws = (float*)d_ws;
  float* d_vec = ws;                          // [B*C]
  float* e_vec = ws + BATCH * C_DIM;          // [B*C]
  float* f_vec = ws + 2 * BATCH * C_DIM;      // [B*C]

  cspfm_pool_kernel<<<BATCH * C_DIM, 256, 0, stream>>>(x, d_vec, e_vec);
  cspfm_attn_kernel<<<BATCH, 512, 0, stream>>>(d_vec, e_vec, alpha, beta, f_vec);
  cspfm_bcast_kernel<<<BATCH * C_DIM, 256, 0, stream>>>(f_vec, out);
}